// GraphConvolution_75127567942080
// MI455X (gfx1250) — compile-verified
//
#include <hip/hip_runtime.h>
#include <math.h>

typedef __attribute__((ext_vector_type(2))) float v2f;
typedef __attribute__((ext_vector_type(8))) float v8f;

#define FEAT 64
#define SSTR 68      // padded LDS row stride (floats); 68 % 4 == 0 keeps float4 stores aligned
#define ALPHA 0.1f
#define LAMDA 0.5f

// ---------------- zero workspace ----------------
__global__ void gcnii_zero(float* __restrict__ p, int n) {
  int i = blockIdx.x * blockDim.x + threadIdx.x;
  if (i < n) p[i] = 0.0f;
}

// ---------------- degree accumulation ----------------
__global__ void gcnii_degrees(const int* __restrict__ src, const int* __restrict__ dst,
                              float* __restrict__ deg_out, float* __restrict__ deg_in, int E) {
  int e = blockIdx.x * blockDim.x + threadIdx.x;
  if (e < E) {
    atomicAdd(&deg_out[src[e]], 1.0f);
    atomicAdd(&deg_in[dst[e]], 1.0f);
  }
}

// ---------------- deg -> deg^{-1/2} in place ----------------
__global__ void gcnii_norms(float* __restrict__ deg_out, float* __restrict__ deg_in, int N) {
  int n = blockIdx.x * blockDim.x + threadIdx.x;
  if (n < N) {
    float a = deg_out[n];
    float b = deg_in[n];
    deg_out[n] = (a > 0.0f) ? rsqrtf(a) : 0.0f;
    deg_in[n]  = (b > 0.0f) ? rsqrtf(b) : 0.0f;
  }
}

// ---------------- edge scatter: agg[dst] += x[src] * norm_src[src] ----------------
// 64 threads per edge -> coalesced full-row gathers; f32 global atomic adds for scatter.
__global__ void gcnii_scatter(const float* __restrict__ x, const int* __restrict__ src,
                              const int* __restrict__ dst, const float* __restrict__ nsrc,
                              float* __restrict__ agg, int ED) {
  int i = blockIdx.x * blockDim.x + threadIdx.x;
  if (i < ED) {
    int e = i >> 6;
    int d = i & 63;
    int s = src[e];
    float v = x[s * FEAT + d] * nsrc[s];
    atomicAdd(&agg[dst[e] * FEAT + d], v);
  }
}

// ---------------- fused: support build + (support @ W) via f32 WMMA + combine ----------------
// 8 waves/block; each wave owns a 16-row tile of the output (16 x 64).
// W (64x64, 16KB) loaded once per block into LDS; per-wave support tile in padded LDS.
// Row-bound guard is hoisted to a wave-uniform branch so full tiles never touch EXEC.
__global__ __launch_bounds__(256)
void gcnii_wmma(const float* __restrict__ x,     // input (also residual)
                const float* __restrict__ h0,
                const float* __restrict__ Wm,    // [64,64] row-major
                const float* __restrict__ agg,   // workspace aggregate
                const float* __restrict__ ndst,  // deg_in^{-1/2}
                const int*   __restrict__ lptr,  // layer index l (device scalar)
                float* __restrict__ out, int N, int ntiles) {
  __shared__ float Wlds[FEAT * FEAT];
  __shared__ float S[8][16 * SSTR];

  int tid = threadIdx.x;

  // cooperative W load (1024 float4 / 256 threads = 4 each)
  {
    const float4* wg = (const float4*)Wm;
    float4* wl = (float4*)Wlds;
    for (int i = tid; i < (FEAT * FEAT) / 4; i += 256) wl[i] = wg[i];
  }
  __syncthreads();

  int wid  = tid >> 5;
  int lane = tid & 31;
  int tile = blockIdx.x * 8 + wid;
  if (tile >= ntiles) return;

  const int rowBase  = tile * 16;
  const bool fullTile = (rowBase + 16 <= N);   // wave-uniform
  float* Sm = &S[wid][0];

  // build support tile: S[r][c] = (1-a)*agg*norm_dst + a*h0   (per-wave, no block sync needed)
  if (fullTile) {
    const float4* aggv = (const float4*)(agg + rowBase * FEAT);
    const float4* h0v  = (const float4*)(h0  + rowBase * FEAT);
    for (int i = lane; i < 16 * FEAT / 4; i += 32) {   // 8 iterations of float4
      int r  = i >> 4;
      int c4 = i & 15;
      float nd = ndst[rowBase + r] * (1.0f - ALPHA);
      float4 av = aggv[i];
      float4 hv = h0v[i];
      float4 sv;
      sv.x = av.x * nd + ALPHA * hv.x;
      sv.y = av.y * nd + ALPHA * hv.y;
      sv.z = av.z * nd + ALPHA * hv.z;
      sv.w = av.w * nd + ALPHA * hv.w;
      *(float4*)&Sm[r * SSTR + c4 * 4] = sv;   // 16B-aligned (SSTR % 4 == 0)
    }
  } else {
    for (int i = lane; i < 16 * FEAT; i += 32) {
      int r = i >> 6;
      int c = i & 63;
      int n = rowBase + r;
      float s = 0.0f;
      if (n < N) s = (1.0f - ALPHA) * agg[n * FEAT + c] * ndst[n] + ALPHA * h0[n * FEAT + c];
      Sm[r * SSTR + c] = s;
    }
  }
  // LDS ops within a wave are in-order; compiler inserts s_wait_dscnt for the loads below.

  int hi = lane >> 4;   // 0|1 -> K half / row half per ISA layout
  int lo = lane & 15;

  float theta = logf(LAMDA / (float)lptr[0] + 1.0f);
  float omt   = 1.0f - theta;

  for (int nt = 0; nt < 4; ++nt) {           // 4 column tiles of 16
    v8f acc = {};
    const int cb = nt * 16;
#pragma unroll
    for (int kb = 0; kb < FEAT; kb += 4) {   // 16 k-steps of K=4
      int k0 = kb + 2 * hi;
      v2f a, b;
      // A 16x4 f32: lane row = lo, VGPR0/1 = K {k0, k0+1}
      a.x = Sm[lo * SSTR + k0];
      a.y = Sm[lo * SSTR + k0 + 1];
      // B 4x16 f32: lane col = lo, VGPR0/1 = K {k0, k0+1}
      b.x = Wlds[k0 * FEAT + cb + lo];
      b.y = Wlds[(k0 + 1) * FEAT + cb + lo];
      acc = __builtin_amdgcn_wmma_f32_16x16x4_f32(
          /*neg_a=*/false, a, /*neg_b=*/false, b,
          /*c_mod=*/(short)0, acc, /*reuse_a=*/false, /*reuse_b=*/false);
    }
    // D layout: VGPR v -> row (v + 8*hi), col = lo
    if (fullTile) {
#pragma unroll
      for (int v = 0; v < 8; ++v) {
        int r   = v + 8 * hi;
        int row = rowBase + r;
        int c   = cb + lo;
        float s = Sm[r * SSTR + c];
        out[row * FEAT + c] = theta * acc[v] + omt * s + x[row * FEAT + c];
      }
    } else {
#pragma unroll
      for (int v = 0; v < 8; ++v) {
        int r   = v + 8 * hi;
        int row = rowBase + r;
        int c   = cb + lo;
        if (row < N) {
          float s = Sm[r * SSTR + c];
          out[row * FEAT + c] = theta * acc[v] + omt * s + x[row * FEAT + c];
        }
      }
    }
  }
}

extern "C" void kernel_launch(void* const* d_in, const int* in_sizes, int n_in,
                              void* d_out, int out_size, void* d_ws, size_t ws_size,
                              hipStream_t stream) {
  const float* x    = (const float*)d_in[0];  // input [N,64]
  const float* h0   = (const float*)d_in[1];  // h0    [N,64]
  const int*   src  = (const int*)  d_in[2];  // [E]
  const int*   dst  = (const int*)  d_in[3];  // [E]
  const float* Wm   = (const float*)d_in[4];  // [64,64]
  const int*   lptr = (const int*)  d_in[5];  // scalar l
  float*       out  = (float*)d_out;

  const int ND = in_sizes[0];
  const int N  = ND / FEAT;
  const int E  = in_sizes[2];

  float* wsf     = (float*)d_ws;
  float* deg_out = wsf;            // N floats -> becomes norm_src
  float* deg_in  = wsf + N;        // N floats -> becomes norm_dst
  float* agg     = wsf + 2 * N;    // N*64 floats

  const int zn = 2 * N + ND;
  gcnii_zero<<<(zn + 255) / 256, 256, 0, stream>>>(wsf, zn);
  gcnii_degrees<<<(E + 255) / 256, 256, 0, stream>>>(src, dst, deg_out, deg_in, E);
  gcnii_norms<<<(N + 255) / 256, 256, 0, stream>>>(deg_out, deg_in, N);

  const int ED = E * FEAT;
  gcnii_scatter<<<(ED + 255) / 256, 256, 0, stream>>>(x, src, dst, deg_out, agg, ED);

  const int ntiles = (N + 15) / 16;
  gcnii_wmma<<<(ntiles + 7) / 8, 256, 0, stream>>>(x, h0, Wm, agg, deg_in, lptr, out, N, ntiles);
}